// ParabolicPool1D_41386304864707
// MI455X (gfx1250) — compile-verified
//
#include <hip/hip_runtime.h>
#include <hip/hip_bf16.h>
#include <stdint.h>

#define KS 7
#define STRIDE 2
#define C 64
#define N 524288
#define HALF_N (N / STRIDE)        // 262144 outputs per channel
#define BLOCK 256
#define OUT_PER_BLOCK 1024         // outputs per block (4 per thread)
#define TILE_IN 2048               // interior inputs per block (2 * OUT_PER_BLOCK)
#define TILE_LDS (TILE_IN + 8)     // + 4-float halo each side

__global__ __launch_bounds__(BLOCK) void parabolic_pool1d_kernel(
    const float* __restrict__ f, const float* __restrict__ t,
    float* __restrict__ out)
{
    __shared__ float tile[TILE_LDS];

    const int tid = threadIdx.x;
    const int c  = blockIdx.y;
    const int X0 = blockIdx.x * OUT_PER_BLOCK;           // first output index of block
    const float* grow = f + (long long)c * N + 2 * X0;   // first interior input (always in range)

    // ---- Async copy of the 2048-float interior f[c, 2*X0 .. 2*X0+2047] -> tile[4..2051].
    // Per thread: two b128 async loads. The instruction offset is added to BOTH the
    // LDS address (VDST vgpr) and the global address (saddr + vaddr), so one LDS-addr
    // VGPR + one byte-offset VGPR serves both loads.
    {
        uint32_t lds_addr = (uint32_t)(uintptr_t)(&tile[4 + 8 * tid]); // low 32 bits = LDS offset
        uint32_t voff     = 32u * (uint32_t)tid;                       // byte offset into row tile
        asm volatile(
            "global_load_async_to_lds_b128 %0, %1, %2\n\t"
            "global_load_async_to_lds_b128 %0, %1, %2 offset:16"
            :
            : "v"(lds_addr), "v"(voff), "s"(grow)
            : "memory");
    }

    // ---- Halos with -inf padding at row edges (guarded scalar loads; never OOB).
    if (tid < 4) {
        // left halo: tile[tid] <- f[c, 2*X0 - 4 + tid]
        int p = 2 * X0 - 4 + tid;
        tile[tid] = (p >= 0) ? f[(long long)c * N + p] : -__builtin_inff();
    } else if (tid < 8) {
        // right halo: tile[2052 + k] <- f[c, 2*X0 + 2048 + k]
        int k = tid - 4;
        int p = 2 * X0 + TILE_IN + k;
        tile[TILE_IN + 4 + k] = (p < N) ? f[(long long)c * N + p] : -__builtin_inff();
    }

    asm volatile("s_wait_asynccnt 0x0" ::: "memory");
    __syncthreads();

    // ---- Parabolic structuring element (symmetric): w(0)=0, w(1)=-inv, w(2)=-4inv, w(3)=-9inv
    const float inv = 0.25f / t[c];   // t > 0 by construction
    const float w1  = -1.0f * inv;
    const float w2  = -4.0f * inv;
    const float w3  = -9.0f * inv;

    // Each thread: 4 consecutive outputs y = X0 + 4*tid + j.
    // Input pos p maps to LDS index p - 2*X0 + 4; center index for output j: 8*tid + 2*j + 4.
    float r0, r1, r2, r3;
    {
        float rr[4];
        #pragma unroll
        for (int j = 0; j < 4; ++j) {
            const int ci = 8 * tid + 2 * j + 4;
            float r = fmaxf(0.0f, tile[ci]);                               // d = 0 (w = 0), clamp at 0
            r = fmaxf(r, fmaxf(tile[ci - 1], tile[ci + 1]) + w1);          // |d| = 1
            r = fmaxf(r, fmaxf(tile[ci - 2], tile[ci + 2]) + w2);          // |d| = 2
            r = fmaxf(r, fmaxf(tile[ci - 3], tile[ci + 3]) + w3);          // |d| = 3
            rr[j] = r;
        }
        r0 = rr[0]; r1 = rr[1]; r2 = rr[2]; r3 = rr[3];
    }

    // Coalesced aligned float4 store.
    float4 res = make_float4(r0, r1, r2, r3);
    float4* op = (float4*)(out + (long long)c * HALF_N + X0 + 4 * tid);
    *op = res;
}

extern "C" void kernel_launch(void* const* d_in, const int* in_sizes, int n_in,
                              void* d_out, int out_size, void* d_ws, size_t ws_size,
                              hipStream_t stream) {
    const float* f = (const float*)d_in[0];  // [C, N] float32
    const float* t = (const float*)d_in[1];  // [C]    float32
    float* out = (float*)d_out;              // [C, N/2] float32
    (void)in_sizes; (void)n_in; (void)out_size; (void)d_ws; (void)ws_size;

    dim3 grid(HALF_N / OUT_PER_BLOCK, C, 1); // (256, 64)
    parabolic_pool1d_kernel<<<grid, BLOCK, 0, stream>>>(f, t, out);
}